// torch_openpose_57019985822572
// MI455X (gfx1250) — compile-verified
//
#include <hip/hip_runtime.h>
#include <math.h>

#define H_ 1080
#define W_ 1920
#define NPART 25
#define NLIMB 26
#define KTOP 32
#define THRE1 0.1f

// Tile geometry: 46x46 outputs -> 48x48 blurred region (3x3 WMMA tiles),
// vertical pass produces 48 rows x 80 cols (5 N-tiles), input patch 72x72
// (padded to width 80 with zeros).
#define OT   46
#define BT   48
#define SINH 72
#define SINW 80
#define SVW  80
#define TAPS 25

typedef float v2f __attribute__((ext_vector_type(2)));
typedef float v8f __attribute__((ext_vector_type(8)));

#if __has_builtin(__builtin_amdgcn_global_load_async_to_lds_b32)
#define USE_ASYNC_LDS 1
#else
#define USE_ASYNC_LDS 0
#endif

// Normalized gaussian taps, sigma=3, radius 12 (matches reference f32 math)
__device__ const float KW[TAPS] = {
  4.46114e-05f, 1.60095e-04f, 5.14107e-04f, 1.477325e-03f, 3.798771e-03f,
  8.740874e-03f, 1.7999499e-02f, 3.3159998e-02f, 5.4671589e-02f, 8.0659196e-02f,
  1.06485687e-01f, 1.25797975e-01f, 1.32984530e-01f, 1.25797975e-01f, 1.06485687e-01f,
  8.0659196e-02f, 5.4671589e-02f, 3.3159998e-02f, 1.7999499e-02f, 8.740874e-03f,
  3.798771e-03f, 1.477325e-03f, 5.14107e-04f, 1.60095e-04f, 4.46114e-05f };

__device__ __forceinline__ int refl(int i, int n) {
  if (i < 0)  i = -1 - i;        // jnp symmetric padding
  if (i >= n) i = 2 * n - 1 - i;
  return i;
}

__global__ void init_counts(int* cnt) {
  if (threadIdx.x < NPART) cnt[threadIdx.x] = 0;
}

__global__ __launch_bounds__(256)
void blur_peaks(const float* __restrict__ heat, int* __restrict__ cnt,
                float* __restrict__ cval, int* __restrict__ cidx, int cap) {
  __shared__ __align__(16) float s_in[SINH * SINW];
  __shared__ __align__(16) float s_v [BT   * SVW ];
  __shared__ __align__(16) float s_h [BT   * BT  ];
  __shared__ float kwp[56];   // padded taps: kwp[i] = KW[i-16] for 16<=i<41 else 0

  const int tid = threadIdx.x;
  const int ch  = blockIdx.z;
  const int gy0 = blockIdx.y * OT;
  const int gx0 = blockIdx.x * OT;

  for (int p = tid; p < 56; p += 256)
    kwp[p] = (p >= 16 && p < 16 + TAPS) ? KW[p - 16] : 0.0f;

  // ---- Stage 72x72 patch (reflect-padded) into LDS; zero-pad cols 72..79
#if USE_ASYNC_LDS
  for (int p = tid; p < SINH * SINW; p += 256) {
    int i = p / SINW, j = p % SINW;
    if (j < SINH) {
      int gy = refl(gy0 - 13 + i, H_);
      int gx = refl(gx0 - 13 + j, W_);
      int* src = (int*)(heat + ((size_t)(gy * W_ + gx) * (NPART + 1) + ch));
      int* dst = (int*)&s_in[p];
      __builtin_amdgcn_global_load_async_to_lds_b32(
          (int __attribute__((address_space(1)))*)src,
          (int __attribute__((address_space(3)))*)dst, 0, 0);
    } else {
      s_in[p] = 0.0f;
    }
  }
  asm volatile("s_wait_asynccnt 0x0" ::: "memory");
#else
  for (int p = tid; p < SINH * SINW; p += 256) {
    int i = p / SINW, j = p % SINW;
    float v = 0.0f;
    if (j < SINH) {
      int gy = refl(gy0 - 13 + i, H_);
      int gx = refl(gx0 - 13 + j, W_);
      v = heat[(size_t)(gy * W_ + gx) * (NPART + 1) + ch];
    }
    s_in[p] = v;
  }
#endif
  __syncthreads();

  const int lane = tid & 31;
  const int wv   = tid >> 5;             // wave id, 8 waves
  const int mn   = lane & 15;            // M (A) or N (B/D) index
  const int hi2  = (lane >> 4) << 1;     // K sub-pair select: 0 or 2
  const int mb   = (lane < 16) ? 0 : 8;  // D-row base

  // Band operand k[q - mn] for K-chunk s: identical for vertical-A and
  // horizontal-B, and for every tile -> hoist into registers once.
  v2f bnd[10];
  #pragma unroll
  for (int s = 0; s < 10; ++s) {
    int base = 4 * s + hi2 - mn + 16;    // in [1,55], always in-bounds
    bnd[s].x = kwp[base];
    bnd[s].y = kwp[base + 1];
  }

  // ---- Vertical blur: Out(16x16) = Band(16x40) x Data(40x16), 10 K=4 steps
  for (int tile = wv; tile < 15; tile += 8) {
    const int rbase = (tile / 5) * 16;
    const int c0    = (tile % 5) * 16;
    const float* bp = &s_in[(rbase + hi2) * SINW + c0 + mn];
    v8f acc = {};
    #pragma unroll
    for (int s = 0; s < 10; ++s) {
      v2f b;
      b.x = bp[s * 4 * SINW];
      b.y = bp[s * 4 * SINW + SINW];
      acc = __builtin_amdgcn_wmma_f32_16x16x4_f32(false, bnd[s], false, b,
                                                  (short)0, acc, false, false);
    }
    #pragma unroll
    for (int v = 0; v < 8; ++v)
      s_v[(rbase + mb + v) * SVW + c0 + mn] = acc[v];
  }
  __syncthreads();

  // ---- Horizontal blur: Out(16x16) = Data(16x40) x Band(40x16)
  for (int tile = wv; tile < 9; tile += 8) {
    const int r0 = (tile / 3) * 16;
    const int c0 = (tile % 3) * 16;
    const v2f* ap = (const v2f*)&s_v[(r0 + mn) * SVW + c0 + hi2];
    v8f acc = {};
    #pragma unroll
    for (int s = 0; s < 10; ++s) {
      v2f a = ap[s * 2];
      acc = __builtin_amdgcn_wmma_f32_16x16x4_f32(false, a, false, bnd[s],
                                                  (short)0, acc, false, false);
    }
    #pragma unroll
    for (int v = 0; v < 8; ++v)
      s_h[(r0 + mb + v) * BT + c0 + mn] = acc[v];
  }
  __syncthreads();

  // ---- Peak detection on 46x46 interior; boundary shifts are zero in ref
  for (int p = tid; p < OT * OT; p += 256) {
    int r = p / OT, c = p % OT;
    int gy = gy0 + r, gx = gx0 + c;
    if (gy >= H_ || gx >= W_) continue;
    float cen = s_h[(r + 1) * BT + (c + 1)];
    bool ok = cen > THRE1;
    ok = ok && (gy == 0      || cen >= s_h[(r    ) * BT + (c + 1)]);
    ok = ok && (gy == H_ - 1 || cen >= s_h[(r + 2) * BT + (c + 1)]);
    ok = ok && (gx == 0      || cen >= s_h[(r + 1) * BT + (c    )]);
    ok = ok && (gx == W_ - 1 || cen >= s_h[(r + 1) * BT + (c + 2)]);
    if (ok) {
      int slot = atomicAdd(&cnt[ch], 1);
      if (slot < cap) {
        cval[(size_t)ch * cap + slot] = s_in[(r + 13) * SINW + (c + 13)];
        cidx[(size_t)ch * cap + slot] = gy * W_ + gx;
      }
    }
  }
}

#define SHCAP 2048

__global__ __launch_bounds__(256)
void select_topk(const int* __restrict__ cnt, const float* __restrict__ cval,
                 const int* __restrict__ cidx, int cap, float* __restrict__ out) {
  __shared__ int   hist[256];
  __shared__ float shv[SHCAP];
  __shared__ int   shi[SHCAP];
  __shared__ float sv[256];
  __shared__ int   si[256];
  __shared__ int   sp[256];
  __shared__ int   scnt;
  __shared__ float thr_s;

  const int ch = blockIdx.x, tid = threadIdx.x;
  int n = cnt[ch]; if (n > cap) n = cap;
  const float* v  = cval + (size_t)ch * cap;
  const int*   ix = cidx + (size_t)ch * cap;

  hist[tid] = 0;
  if (tid == 0) scnt = 0;
  __syncthreads();

  // Pass 1: histogram of candidate values (original map values in [0,1))
  for (int i = tid; i < n; i += 256) {
    int b = (int)(v[i] * 256.0f);
    b = b < 0 ? 0 : (b > 255 ? 255 : b);
    atomicAdd(&hist[b], 1);
  }
  __syncthreads();
  if (tid == 0) {
    int acc = 0, bc = 0;
    for (int b = 255; b >= 0; --b) {
      acc += hist[b];
      if (acc >= KTOP) { bc = b; break; }
    }
    thr_s = (float)bc * (1.0f / 256.0f);
  }
  __syncthreads();
  const float thr = thr_s;

  // Pass 2: compact candidates above threshold into LDS shortlist
  for (int i = tid; i < n; i += 256) {
    float vv = v[i];
    if (vv >= thr) {
      int s = atomicAdd(&scnt, 1);
      if (s < SHCAP) { shv[s] = vv; shi[s] = ix[i]; }
    }
  }
  __syncthreads();
  int m = scnt; if (m > SHCAP) m = SHCAP;

  // 32 argmax passes over the small shortlist (ties: smaller flat index wins)
  for (int k = 0; k < KTOP; ++k) {
    float bv = -3.0e38f; int bi = 0x7fffffff, bp = -1;
    for (int i = tid; i < m; i += 256) {
      float vv = shv[i]; int ii = shi[i];
      if (vv > bv || (vv == bv && ii < bi)) { bv = vv; bi = ii; bp = i; }
    }
    sv[tid] = bv; si[tid] = bi; sp[tid] = bp;
    __syncthreads();
    for (int o = 128; o > 0; o >>= 1) {
      if (tid < o) {
        if (sv[tid + o] > sv[tid] ||
            (sv[tid + o] == sv[tid] && si[tid + o] < si[tid])) {
          sv[tid] = sv[tid + o]; si[tid] = si[tid + o]; sp[tid] = sp[tid + o];
        }
      }
      __syncthreads();
    }
    if (tid == 0) {
      float px = 0.f, py = 0.f, ps = 0.f, mk = 0.f;
      if (sp[0] >= 0 && sv[0] > -1.0e30f) {
        int fi = si[0];
        px = (float)(fi % W_);
        py = (float)(fi / W_);
        ps = sv[0]; mk = 1.0f;
        shv[sp[0]] = -3.0e38f;   // remove selected candidate
      }
      out[                   ch * KTOP + k] = px;
      out[1 * NPART * KTOP + ch * KTOP + k] = py;
      out[2 * NPART * KTOP + ch * KTOP + k] = ps;
      out[3 * NPART * KTOP + ch * KTOP + k] = mk;
    }
    __syncthreads();
  }
}

__device__ const int c_limb[NLIMB][2] = {
  {1,8},{1,2},{1,5},{2,3},{3,4},{5,6},{6,7},{8,9},{9,10},{10,11},
  {8,12},{12,13},{13,14},{1,0},{0,15},{15,16},{0,17},{0,18},{14,19},
  {19,20},{14,21},{11,22},{22,23},{11,24},{2,17},{5,18}};

__global__ __launch_bounds__(256)
void conn_kernel(const float* __restrict__ paf, const float* __restrict__ pk,
                 float* __restrict__ conn) {
  const int l = blockIdx.x, tid = threadIdx.x;
  __shared__ float ax[KTOP], ay[KTOP], bx[KTOP], by[KTOP];
  __shared__ int   ma[KTOP], mb[KTOP];
  const int la = c_limb[l][0], lb = c_limb[l][1];
  if (tid < KTOP) {
    ax[tid] = pk[                   la * KTOP + tid];
    ay[tid] = pk[1 * NPART * KTOP + la * KTOP + tid];
    ma[tid] = pk[3 * NPART * KTOP + la * KTOP + tid] > 0.5f;
    bx[tid] = pk[                   lb * KTOP + tid];
    by[tid] = pk[1 * NPART * KTOP + lb * KTOP + tid];
    mb[tid] = pk[3 * NPART * KTOP + lb * KTOP + tid] > 0.5f;
  }
  __syncthreads();
  for (int p = tid; p < KTOP * KTOP; p += 256) {
    int i = p >> 5, j = p & 31;               // i: A peak, j: B peak
    float vx = bx[j] - ax[i], vy = by[j] - ay[i];
    float nrm = sqrtf(vx * vx + vy * vy) + 1e-10f;
    float ux = vx / nrm, uy = vy / nrm;
    float s = 0.f;
    #pragma unroll
    for (int q = 0; q < 10; ++q) {
      float t = (float)q * (1.0f / 9.0f);
      int sx = (int)rintf(ax[i] + vx * t);    // round-half-even == jnp.round
      int sy = (int)rintf(ay[i] + vy * t);
      int base = (sy * W_ + sx) * (2 * NLIMB) + (l << 1);
      s += paf[base] * ux + paf[base + 1] * uy;
    }
    s *= 0.1f;
    s = fminf(fmaxf(s, 0.f), 1.f);
    conn[l * KTOP * KTOP + p] = (ma[i] && mb[j]) ? s : 0.f;
  }
}

extern "C" void kernel_launch(void* const* d_in, const int* in_sizes, int n_in,
                              void* d_out, int out_size, void* d_ws, size_t ws_size,
                              hipStream_t stream) {
  const float* heat = (const float*)d_in[0];
  const float* paf  = (const float*)d_in[1];
  float* out = (float*)d_out;

  long cap_l = ((long)ws_size - 128) / (NPART * 8);
  int cap = cap_l > 131072 ? 131072 : (cap_l < 256 ? 256 : (int)cap_l);

  char* ws   = (char*)d_ws;
  int*   cnt = (int*)ws;
  float* cvl = (float*)(ws + 128);
  int*   cix = (int*)(ws + 128 + (size_t)NPART * cap * 4);

  init_counts<<<1, 32, 0, stream>>>(cnt);
  dim3 grid((W_ + OT - 1) / OT, (H_ + OT - 1) / OT, NPART);
  blur_peaks<<<grid, 256, 0, stream>>>(heat, cnt, cvl, cix, cap);
  select_topk<<<NPART, 256, 0, stream>>>(cnt, cvl, cix, cap, out);
  conn_kernel<<<NLIMB, 256, 0, stream>>>(paf, out, out + 4 * NPART * KTOP);
}